// SREXmodel_75445395521552
// MI455X (gfx1250) — compile-verified
//
#include <hip/hip_runtime.h>
#include <math.h>

#define NN 40000
#define EE 400000
#define FF 16
#define HH 8
#define CC 64
#define HC 512          // H*C
#define RR 40
#define NOUT 61         // MAXR+1
#define NLOG 39         // min(R1,R2)-1
#define SLOPE 0.2f

typedef float v2f __attribute__((ext_vector_type(2)));
typedef float v8f __attribute__((ext_vector_type(8)));

// ---------------- h = x @ W_gat  (40000x16 @ 16x512) via V_WMMA_F32_16X16X4_F32
// grid = (2500, 8), block = 128 (4 waves); each wave owns one 16x16 D tile.
__global__ void k_gemm(const float* __restrict__ x, const float* __restrict__ W,
                       float* __restrict__ hbuf) {
  int lane  = threadIdx.x & 31;
  int wave  = threadIdx.x >> 5;
  int ntile = blockIdx.y * 4 + wave;      // 0..31
  int mbase = blockIdx.x * 16;            // 0..39984
  int half  = lane >> 4;                  // 0: K lo pair, 1: K hi pair
  int l15   = lane & 15;
  int col   = ntile * 16 + l15;
  int row   = mbase + l15;
  v8f c = {};
#pragma unroll
  for (int s = 0; s < 4; ++s) {
    int kb = s * 4 + half * 2;            // A layout: lanes0-15 K={0,1}, lanes16-31 K={2,3}
    v2f a, b;
    a.x = x[row * FF + kb];
    a.y = x[row * FF + kb + 1];
    b.x = W[kb * HC + col];
    b.y = W[(kb + 1) * HC + col];
    c = __builtin_amdgcn_wmma_f32_16x16x4_f32(false, a, false, b, (short)0, c,
                                              false, false);
  }
#pragma unroll
  for (int r = 0; r < 8; ++r) {           // C/D: VGPR r -> M=r (+8 for upper half-wave)
    int m = mbase + half * 8 + r;
    hbuf[(size_t)m * HC + col] = c[r];
  }
}

// ---------------- a_s / a_d per (node, head)
__global__ void k_att(const float* __restrict__ hbuf, const float* __restrict__ att_s,
                      const float* __restrict__ att_d, float* __restrict__ as,
                      float* __restrict__ ad) {
  int id = blockIdx.x * blockDim.x + threadIdx.x;
  if (id >= NN * HH) return;
  int n = id >> 3, hh = id & 7;
  const float* hp = hbuf + (size_t)n * HC + hh * CC;
  const float* ps = att_s + hh * CC;
  const float* pd = att_d + hh * CC;
  float s = 0.f, d = 0.f;
  for (int c = 0; c < CC; ++c) { float v = hp[c]; s += v * ps[c]; d += v * pd[c]; }
  as[id] = s;
  ad[id] = d;
}

// ---------------- CSR-by-dst build
__global__ void k_count(const int* __restrict__ ei, int* __restrict__ counts) {
  int e = blockIdx.x * blockDim.x + threadIdx.x;
  if (e >= EE) return;
  atomicAdd(&counts[ei[EE + e]], 1);
}

__global__ void k_scan(const int* __restrict__ counts, int* __restrict__ offs,
                       int* __restrict__ cursor) {
  __shared__ int ssum[1024];
  int tid = threadIdx.x;
  const int per = (NN + 1023) / 1024;
  int start = tid * per;
  int s = 0;
  for (int i = 0; i < per; ++i) { int idx = start + i; if (idx < NN) s += counts[idx]; }
  ssum[tid] = s;
  __syncthreads();
  if (tid == 0) {
    int acc = 0;
    for (int t = 0; t < 1024; ++t) { int v = ssum[t]; ssum[t] = acc; acc += v; }
    offs[NN] = acc;
  }
  __syncthreads();
  int acc = ssum[tid];
  for (int i = 0; i < per; ++i) {
    int idx = start + i;
    if (idx < NN) { offs[idx] = acc; cursor[idx] = acc; acc += counts[idx]; }
  }
}

__global__ void k_fill(const int* __restrict__ ei, int* __restrict__ cursor,
                       int* __restrict__ eids) {
  int e = blockIdx.x * blockDim.x + threadIdx.x;
  if (e >= EE) return;
  int d = ei[EE + e];
  int pos = atomicAdd(&cursor[d], 1);
  eids[pos] = e;
}

// ---------------- per-dst softmax + aggregation + route reduce (one wave per node)
__global__ void k_node(const float* __restrict__ hbuf, const float* __restrict__ as,
                       const float* __restrict__ ad, const int* __restrict__ offs,
                       const int* __restrict__ eids, const int* __restrict__ ei,
                       const int* __restrict__ route, const float* __restrict__ bias,
                       float* __restrict__ agg) {
  int n = blockIdx.x * 8 + (threadIdx.x >> 5);
  int lane = threadIdx.x & 31;
  if (n >= NN) return;
  int beg = offs[n], end = offs[n + 1];

  float adn[HH], m[HH];
#pragma unroll
  for (int h = 0; h < HH; ++h) {
    adn[h] = ad[n * HH + h];
    float e = as[n * HH + h] + adn[h];          // self loop edge
    m[h] = e < 0.f ? e * SLOPE : e;
  }
  // pass 1: segment max (every lane redundantly, broadcast loads)
  for (int j = beg; j < end; ++j) {
    int s = ei[eids[j]];
#pragma unroll
    for (int h = 0; h < HH; ++h) {
      float e = as[s * HH + h] + adn[h];
      e = e < 0.f ? e * SLOPE : e;
      m[h] = fmaxf(m[h], e);
    }
  }
  // pass 2: denominator
  float den[HH], aself[HH];
#pragma unroll
  for (int h = 0; h < HH; ++h) {
    float e = as[n * HH + h] + adn[h];
    e = e < 0.f ? e * SLOPE : e;
    aself[h] = __expf(e - m[h]);
    den[h] = aself[h];
  }
  for (int j = beg; j < end; ++j) {
    int s = ei[eids[j]];
#pragma unroll
    for (int h = 0; h < HH; ++h) {
      float e = as[s * HH + h] + adn[h];
      e = e < 0.f ? e * SLOPE : e;
      den[h] += __expf(e - m[h]);
    }
  }
  // pass 3: weighted aggregation of h[src]; lane owns channels lane+32*i
  float acc[16];
  const float* hn = hbuf + (size_t)n * HC;
#pragma unroll
  for (int i = 0; i < 16; ++i) {
    int c = lane + 32 * i;
    acc[i] = hn[c] * (aself[c >> 6] / den[c >> 6]);
  }
  for (int j = beg; j < end; ++j) {
    int s = ei[eids[j]];
    if (j + 1 < end) __builtin_prefetch(hbuf + (size_t)ei[eids[j + 1]] * HC, 0, 1);
    float alf[HH];
#pragma unroll
    for (int h = 0; h < HH; ++h) {
      float e = as[s * HH + h] + adn[h];
      e = e < 0.f ? e * SLOPE : e;
      alf[h] = __expf(e - m[h]) / den[h];
    }
    const float* hs = hbuf + (size_t)s * HC;
#pragma unroll
    for (int i = 0; i < 16; ++i) {
      int c = lane + 32 * i;
      acc[i] += hs[c] * alf[c >> 6];
    }
  }
  // emb = acc + bias_gat; fold directly into route aggregate (L2-resident 80KB)
  int r = route[n];
#pragma unroll
  for (int i = 0; i < 16; ++i) {
    int c = lane + 32 * i;
    atomicAdd(&agg[r * HC + c], acc[i] + bias[c]);
  }
}

// ---------------- head: P1[i,k] = agg1[i]·Wh[:512,k] + b[k]; P2[j,k] = agg2[j]·Wh[512:,k]
__global__ void k_head(const float* __restrict__ agg1, const float* __restrict__ agg2,
                       const float* __restrict__ Wh, const float* __restrict__ bh,
                       float* __restrict__ P) {
  int id = blockIdx.x * blockDim.x + threadIdx.x;
  if (id >= 2 * RR * NOUT) return;
  int side = id / (RR * NOUT);
  int rem = id - side * RR * NOUT;
  int i = rem / NOUT, k = rem - i * NOUT;
  const float* A = (side ? agg2 : agg1) + i * HC;
  const float* W = Wh + (size_t)side * HC * NOUT + k;
  float s = side ? 0.f : bh[k];
  for (int d = 0; d < HC; ++d) s += A[d] * W[(size_t)d * NOUT];
  P[id] = s;
}

// ---------------- global softmax over 40*40*39 logits, single workgroup
__global__ void k_softmax(const float* __restrict__ P, float* __restrict__ out) {
  __shared__ float red[1024];
  const float* P1 = P;
  const float* P2 = P + RR * NOUT;
  int tid = threadIdx.x;
  const int T = RR * RR * NLOG;
  float mx = -INFINITY;
  for (int t = tid; t < T; t += 1024) {
    int i = t / (RR * NLOG);
    int rem = t - i * (RR * NLOG);
    int j = rem / NLOG, k = rem - j * NLOG;
    float L = P1[i * NOUT + k] + P2[j * NOUT + k];
    mx = fmaxf(mx, L);
  }
  red[tid] = mx;
  __syncthreads();
  for (int s = 512; s > 0; s >>= 1) {
    if (tid < s) red[tid] = fmaxf(red[tid], red[tid + s]);
    __syncthreads();
  }
  float M = red[0];
  __syncthreads();
  float sum = 0.f;
  for (int t = tid; t < T; t += 1024) {
    int i = t / (RR * NLOG);
    int rem = t - i * (RR * NLOG);
    int j = rem / NLOG, k = rem - j * NLOG;
    float L = P1[i * NOUT + k] + P2[j * NOUT + k];
    sum += __expf(L - M);
  }
  red[tid] = sum;
  __syncthreads();
  for (int s = 512; s > 0; s >>= 1) {
    if (tid < s) red[tid] += red[tid + s];
    __syncthreads();
  }
  float S = red[0];
  for (int t = tid; t < T; t += 1024) {
    int i = t / (RR * NLOG);
    int rem = t - i * (RR * NLOG);
    int j = rem / NLOG, k = rem - j * NLOG;
    float L = P1[i * NOUT + k] + P2[j * NOUT + k];
    out[t] = __expf(L - M) / S;
  }
}

extern "C" void kernel_launch(void* const* d_in, const int* in_sizes, int n_in,
                              void* d_out, int out_size, void* d_ws, size_t ws_size,
                              hipStream_t stream) {
  (void)in_sizes; (void)n_in; (void)out_size; (void)ws_size;
  const float* x[2]   = {(const float*)d_in[0], (const float*)d_in[5]};
  const int* ei[2]    = {(const int*)d_in[1], (const int*)d_in[6]};
  const int* route[2] = {(const int*)d_in[3], (const int*)d_in[8]};
  const float* Wg   = (const float*)d_in[10];
  const float* asrc = (const float*)d_in[11];
  const float* adst = (const float*)d_in[12];
  const float* bg   = (const float*)d_in[13];
  const float* Wh   = (const float*)d_in[14];
  const float* bh   = (const float*)d_in[15];
  float* out = (float*)d_out;

  char* base = (char*)d_ws;
  size_t off = 0;
  auto take = [&](size_t bytes) -> void* {
    void* r = base + off;
    off = (off + bytes + 255) & ~(size_t)255;
    return r;
  };
  float* hbuf = (float*)take((size_t)NN * HC * 4);   // 81.9 MB, reused per graph
  float* asb  = (float*)take((size_t)NN * HH * 4);
  float* adb  = (float*)take((size_t)NN * HH * 4);
  int* counts = (int*)take((size_t)NN * 4);
  int* offs   = (int*)take((size_t)(NN + 1) * 4);
  int* cursor = (int*)take((size_t)NN * 4);
  int* eidb   = (int*)take((size_t)EE * 4);
  float* agg0 = (float*)take((size_t)RR * HC * 4);
  float* agg1 = (float*)take((size_t)RR * HC * 4);
  float* P    = (float*)take((size_t)2 * RR * NOUT * 4);
  float* aggs[2] = {agg0, agg1};

  hipMemsetAsync(agg0, 0, (size_t)RR * HC * 4, stream);
  hipMemsetAsync(agg1, 0, (size_t)RR * HC * 4, stream);

  for (int g = 0; g < 2; ++g) {
    k_gemm<<<dim3(NN / 16, HC / 64), 128, 0, stream>>>(x[g], Wg, hbuf);
    k_att<<<(NN * HH + 255) / 256, 256, 0, stream>>>(hbuf, asrc, adst, asb, adb);
    hipMemsetAsync(counts, 0, (size_t)NN * 4, stream);
    k_count<<<(EE + 255) / 256, 256, 0, stream>>>(ei[g], counts);
    k_scan<<<1, 1024, 0, stream>>>(counts, offs, cursor);
    k_fill<<<(EE + 255) / 256, 256, 0, stream>>>(ei[g], cursor, eidb);
    k_node<<<NN / 8, 256, 0, stream>>>(hbuf, asb, adb, offs, eidb, ei[g], route[g],
                                       bg, aggs[g]);
  }
  k_head<<<(2 * RR * NOUT + 255) / 256, 256, 0, stream>>>(agg0, agg1, Wh, bh, P);
  k_softmax<<<1, 1024, 0, stream>>>(P, out);
}